// GAT_77953656422757
// MI455X (gfx1250) — compile-verified
//
#include <hip/hip_runtime.h>
#include <hip/hip_bf16.h>

// ---------------- problem constants (match reference) ----------------
constexpr int NTN   = 65536;   // total nodes
constexpr int NPERG = 2048;    // nodes per graph
constexpr int BGR   = 32;      // graphs
constexpr int NFEAT = 128;     // input features
constexpr int NCLS  = 10;      // classes
constexpr float NEGV = -1e9f;

typedef __attribute__((ext_vector_type(2))) float v2f;
typedef __attribute__((ext_vector_type(8))) float v8f;

// ---------------- helpers ----------------
__device__ __forceinline__ unsigned f2key(float f) {
  unsigned b = __float_as_uint(f);
  return (b & 0x80000000u) ? ~b : (b | 0x80000000u);
}
__device__ __forceinline__ float key2f(unsigned k) {
  unsigned b = (k & 0x80000000u) ? (k & 0x7FFFFFFFu) : ~k;
  return __uint_as_float(b);
}

// ---------------- batchnorm statistics (biased var) ----------------
__global__ void bn_stats_kernel(const float* __restrict__ x,
                                float* __restrict__ mu, float* __restrict__ rstd) {
  __shared__ double ss[256], ss2[256];
  int c = blockIdx.x;                       // feature column 0..127
  double s = 0.0, s2 = 0.0;
  for (int n = threadIdx.x; n < NTN; n += blockDim.x) {
    float v = x[(size_t)n * NFEAT + c];
    s += v; s2 += (double)v * (double)v;
  }
  ss[threadIdx.x] = s; ss2[threadIdx.x] = s2;
  __syncthreads();
  for (int st = 128; st > 0; st >>= 1) {
    if ((int)threadIdx.x < st) { ss[threadIdx.x] += ss[threadIdx.x + st]; ss2[threadIdx.x] += ss2[threadIdx.x + st]; }
    __syncthreads();
  }
  if (threadIdx.x == 0) {
    double mean = ss[0] / (double)NTN;
    double var  = ss2[0] / (double)NTN - mean * mean;
    mu[c]   = (float)mean;
    rstd[c] = (float)(1.0 / sqrt(var + 1e-5));
  }
}

// fold batchnorm into W1:  h1 = xn@W1 = x@W1f + brow
__global__ void fold_w1_kernel(const float* __restrict__ W1, const float* __restrict__ gamma,
                               const float* __restrict__ beta, const float* __restrict__ mu,
                               const float* __restrict__ rstd,
                               float* __restrict__ W1f, float* __restrict__ brow) {
  for (int i = threadIdx.x; i < NFEAT * 48; i += blockDim.x) {
    int k = i / 48;
    W1f[i] = W1[i] * gamma[k] * rstd[k];
  }
  if (threadIdx.x < 48) {
    float s = 0.f;
    for (int k = 0; k < NFEAT; ++k)
      s += (beta[k] - mu[k] * gamma[k] * rstd[k]) * W1[k * 48 + threadIdx.x];
    brow[threadIdx.x] = s;
  }
}

// ---------------- f32 WMMA GEMM:  C[M,NACT] = A[M,K] @ B[K,NACT] (+rowbias) ----------------
// B staged in LDS as (B[2p][n], B[2p+1][n]) pairs -> one ds_load_b64 per fragment.
// One wave per 16-row tile; NTILE column tiles of 16; K stepped by 4 (V_WMMA_F32_16X16X4_F32).
template<int K, int NACT>
__global__ __launch_bounds__(256) void wmma_gemm_kernel(
    const float* __restrict__ A, const float* __restrict__ Bm,
    float* __restrict__ C, const float* __restrict__ rowbias) {
  constexpr int NTILE = (NACT + 15) / 16;
  constexpr int NPAD  = NTILE * 16;
  __shared__ v2f ldsB[(K / 2) * NPAD];

  // cooperative stage of B into LDS (packed row-pairs)
  for (int i = threadIdx.x; i < (K / 2) * NPAD; i += blockDim.x) {
    int p = i / NPAD, n = i % NPAD;
    v2f val;
    if (NACT == NPAD || n < NACT) {
      val.x = Bm[(size_t)(2 * p)     * NACT + n];
      val.y = Bm[(size_t)(2 * p + 1) * NACT + n];
    } else { val.x = 0.f; val.y = 0.f; }
    ldsB[i] = val;
  }
  __syncthreads();

  int wave = threadIdx.x >> 5;
  int lane = threadIdx.x & 31;
  int tile = blockIdx.x * (blockDim.x >> 5) + wave;
  int row0 = tile * 16;
  if (row0 >= NTN) return;
  int m  = lane & 15;      // row within A-frag / column within B,C frags
  int kh = lane >> 4;      // which K half-pair this lane holds

  v8f acc[NTILE];
#pragma unroll
  for (int nt = 0; nt < NTILE; ++nt) {
    float bias = 0.f;
    int col = nt * 16 + m;
    if (rowbias != nullptr && (NACT == NPAD || col < NACT)) bias = rowbias[col];
#pragma unroll
    for (int j = 0; j < 8; ++j) acc[nt][j] = bias;
  }

  const float* arow = A + (size_t)(row0 + m) * K;
#pragma unroll 4
  for (int k = 0; k < K; k += 4) {
    // A frag (16x4 f32): VGPR0 = K {0|2}, VGPR1 = K {1|3} split by lane half
    v2f a = *(const v2f*)(arow + k + 2 * kh);
    __builtin_prefetch(arow + k + 64, 0, 1);     // gfx1250 global_prefetch_b8
#pragma unroll
    for (int nt = 0; nt < NTILE; ++nt) {
      v2f b = ldsB[((k >> 1) + kh) * NPAD + nt * 16 + m];
      acc[nt] = __builtin_amdgcn_wmma_f32_16x16x4_f32(
          false, a, false, b, (short)0, acc[nt], false, false);
    }
  }

#pragma unroll
  for (int nt = 0; nt < NTILE; ++nt) {
    int col = nt * 16 + m;
    if (NACT == NPAD || col < NACT) {
#pragma unroll
      for (int j = 0; j < 8; ++j) {
        int r = row0 + j + 8 * kh;       // C/D: VGPR j holds rows j (lanes 0-15) / j+8 (16-31)
        C[(size_t)r * NACT + col] = acc[nt][j];
      }
    }
  }
}

// ---------------- attention logits  als/ald[n,h] = sum_c h[n,h,c]*a[h,c] ----------------
__global__ void attn_logits_kernel(const float* __restrict__ h,
                                   const float* __restrict__ asrc, const float* __restrict__ adst,
                                   float* __restrict__ als, float* __restrict__ ald,
                                   int H, int C, int total) {
  int i = blockIdx.x * blockDim.x + threadIdx.x;
  if (i >= total) return;
  int n = i / H, hh = i % H;
  const float* hr = h + (size_t)n * (H * C) + hh * C;
  float s = 0.f, d = 0.f;
  for (int c = 0; c < C; ++c) { float v = hr[c]; s += v * asrc[hh * C + c]; d += v * adst[hh * C + c]; }
  als[i] = s; ald[i] = d;
}

// ---------------- fills / bias-init ----------------
__global__ void fill_f32_kernel(float* p, float v, int n) {
  int i = blockIdx.x * blockDim.x + threadIdx.x; if (i < n) p[i] = v;
}
__global__ void fill_u32_kernel(unsigned* p, unsigned v, int n) {
  int i = blockIdx.x * blockDim.x + threadIdx.x; if (i < n) p[i] = v;
}
__global__ void init_bias_kernel(float* __restrict__ out, const float* __restrict__ b, int F, int total) {
  int i = blockIdx.x * blockDim.x + threadIdx.x;
  if (i < total) out[i] = b[i % F];
}

// ---------------- edge passes ----------------
template<int H>
__global__ void edge_pass1_kernel(const int* __restrict__ ei, int E, int ETOT,
                                  const int* __restrict__ nm,
                                  const float* __restrict__ als, const float* __restrict__ ald,
                                  float* __restrict__ ebuf, unsigned* __restrict__ mkey) {
  int e = blockIdx.x * blockDim.x + threadIdx.x;
  if (e >= ETOT) return;
  int s, d;
  if (e < E) { s = ei[e]; d = ei[E + e]; } else { s = d = e - E; }
  bool ok = (nm == nullptr) || (nm[s] != 0 && nm[d] != 0);
#pragma unroll
  for (int hh = 0; hh < H; ++hh) {
    float v = NEGV;
    if (ok) {
      v = als[s * H + hh] + ald[d * H + hh];
      v = (v >= 0.f) ? v : 0.2f * v;                 // leaky_relu 0.2
      atomicMax(&mkey[d * H + hh], f2key(v));        // segment max (ordered-key u32 atomic)
    }
    ebuf[(size_t)e * H + hh] = v;
  }
}

template<int H>
__global__ void edge_pass2_kernel(const int* __restrict__ ei, int E, int ETOT,
                                  const int* __restrict__ nm,
                                  float* __restrict__ ebuf, const unsigned* __restrict__ mkey,
                                  float* __restrict__ den) {
  int e = blockIdx.x * blockDim.x + threadIdx.x;
  if (e >= ETOT) return;
  int s, d;
  if (e < E) { s = ei[e]; d = ei[E + e]; } else { s = d = e - E; }
  bool ok = (nm == nullptr) || (nm[s] != 0 && nm[d] != 0);
#pragma unroll
  for (int hh = 0; hh < H; ++hh) {
    float ex = 0.f;
    if (ok) {
      float mx = key2f(mkey[d * H + hh]);            // >= NEGV by construction
      float v  = ebuf[(size_t)e * H + hh];
      ex = __expf(v - mx);
      atomicAdd(&den[d * H + hh], ex);
    }
    ebuf[(size_t)e * H + hh] = ex;
  }
}

template<int H, int C>
__global__ void edge_pass3_kernel(const int* __restrict__ ei, int E, int ETOT,
                                  const float* __restrict__ ebuf, const float* __restrict__ den,
                                  const float* __restrict__ hmat, float* __restrict__ out) {
  int e = blockIdx.x * blockDim.x + threadIdx.x;
  if (e >= ETOT) return;
  int s, d;
  if (e < E) { s = ei[e]; d = ei[E + e]; } else { s = d = e - E; }
#pragma unroll
  for (int hh = 0; hh < H; ++hh) {
    float ex = ebuf[(size_t)e * H + hh];
    if (ex != 0.f) {
      float alpha = ex / fmaxf(den[d * H + hh], 1e-16f);
      const float* hs = hmat + (size_t)s * (H * C) + hh * C;
      float* od = out + (size_t)d * (H * C) + hh * C;
      for (int c = 0; c < C; ++c) atomicAdd(&od[c], alpha * hs[c]);
    }
  }
}

// ---------------- vector norm ----------------
__global__ void vec_norm_kernel(const float* __restrict__ p, int len, float* __restrict__ out) {
  __shared__ float s[64];
  float a = 0.f;
  for (int i = threadIdx.x; i < len; i += 64) a += p[i] * p[i];
  s[threadIdx.x] = a; __syncthreads();
  for (int st = 32; st > 0; st >>= 1) {
    if ((int)threadIdx.x < st) s[threadIdx.x] += s[threadIdx.x + st];
    __syncthreads();
  }
  if (threadIdx.x == 0) out[0] = sqrtf(s[0]);
}

// ---------------- top-k pool: one workgroup per graph, bitonic sort in LDS ----------------
template<int F>
__global__ __launch_bounds__(1024) void topk_pool_kernel(
    const float* __restrict__ xi, const float* __restrict__ p, const float* __restrict__ pnorm,
    const int* __restrict__ nm_in, int* __restrict__ nm_out,
    float* __restrict__ xo, int kkeep) {
  __shared__ float key[NPERG];
  __shared__ int   sidx[NPERG];
  __shared__ float sc[NPERG];
  __shared__ unsigned char sel[NPERG];
  int g = blockIdx.x;
  int base = g * NPERG;
  float invn = 1.f / pnorm[0];

  for (int n = threadIdx.x; n < NPERG; n += blockDim.x) {
    const float* xr = xi + (size_t)(base + n) * F;
    float s = 0.f;
    for (int f = 0; f < F; ++f) s += xr[f] * p[f];
    s *= invn;
    sc[n] = s;
    bool valid = (nm_in == nullptr) || (nm_in[base + n] != 0);
    key[n]  = valid ? s : NEGV;
    sidx[n] = n;
    sel[n]  = 0;
  }
  __syncthreads();

  // bitonic sort, descending
  for (int kk = 2; kk <= NPERG; kk <<= 1) {
    for (int j = kk >> 1; j > 0; j >>= 1) {
      for (int t = threadIdx.x; t < NPERG; t += blockDim.x) {
        int ixj = t ^ j;
        if (ixj > t) {
          bool up = ((t & kk) == 0);       // region direction
          float a = key[t], b = key[ixj];
          bool sw = up ? (a < b) : (a > b);
          if (sw) {
            key[t] = b; key[ixj] = a;
            int tmp = sidx[t]; sidx[t] = sidx[ixj]; sidx[ixj] = tmp;
          }
        }
      }
      __syncthreads();
    }
  }

  for (int t = threadIdx.x; t < NPERG; t += blockDim.x)
    if (t < kkeep) sel[sidx[t]] = 1;
  __syncthreads();

  for (int n = threadIdx.x; n < NPERG; n += blockDim.x) {
    bool valid = (nm_in == nullptr) || (nm_in[base + n] != 0);
    bool keep = (sel[n] != 0) && valid;
    nm_out[base + n] = keep ? 1 : 0;
    float scale = keep ? tanhf(sc[n]) : 0.f;
    const float* xr = xi + (size_t)(base + n) * F;
    float* xw = xo + (size_t)(base + n) * F;
    for (int f = 0; f < F; ++f) xw[f] = xr[f] * scale;
  }
}

// ---------------- final mean-pool (/512) + log_softmax ----------------
__global__ void pooled_lsm_kernel(const float* __restrict__ x, float* __restrict__ out) {
  __shared__ float acc[256 * NCLS];
  int g = blockIdx.x;
  float loc[NCLS];
#pragma unroll
  for (int f = 0; f < NCLS; ++f) loc[f] = 0.f;
  for (int n = threadIdx.x; n < NPERG; n += blockDim.x) {
    const float* xr = x + (size_t)(g * NPERG + n) * NCLS;
#pragma unroll
    for (int f = 0; f < NCLS; ++f) loc[f] += xr[f];
  }
#pragma unroll
  for (int f = 0; f < NCLS; ++f) acc[threadIdx.x * NCLS + f] = loc[f];
  __syncthreads();
  for (int st = 128; st > 0; st >>= 1) {
    if ((int)threadIdx.x < st)
#pragma unroll
      for (int f = 0; f < NCLS; ++f)
        acc[threadIdx.x * NCLS + f] += acc[(threadIdx.x + st) * NCLS + f];
    __syncthreads();
  }
  if (threadIdx.x == 0) {
    float v[NCLS], mx = -3.4e38f;
#pragma unroll
    for (int f = 0; f < NCLS; ++f) { v[f] = acc[f] / 512.f; mx = fmaxf(mx, v[f]); }
    float se = 0.f;
#pragma unroll
    for (int f = 0; f < NCLS; ++f) se += __expf(v[f] - mx);
    float lse = __logf(se) + mx;
#pragma unroll
    for (int f = 0; f < NCLS; ++f) out[g * NCLS + f] = v[f] - lse;
  }
}

// ---------------- launcher ----------------
extern "C" void kernel_launch(void* const* d_in, const int* in_sizes, int n_in,
                              void* d_out, int out_size, void* d_ws, size_t ws_size,
                              hipStream_t stream) {
  (void)n_in; (void)out_size; (void)ws_size;
  const float* x     = (const float*)d_in[0];
  const int*   ei    = (const int*)d_in[1];
  const float* gamma = (const float*)d_in[3];
  const float* beta  = (const float*)d_in[4];
  const float* W1 = (const float*)d_in[5];  const float* as1 = (const float*)d_in[6];
  const float* ad1 = (const float*)d_in[7]; const float* b1 = (const float*)d_in[8];
  const float* W2 = (const float*)d_in[9];  const float* as2 = (const float*)d_in[10];
  const float* ad2 = (const float*)d_in[11]; const float* b2 = (const float*)d_in[12];
  const float* W3 = (const float*)d_in[13]; const float* as3 = (const float*)d_in[14];
  const float* ad3 = (const float*)d_in[15]; const float* b3 = (const float*)d_in[16];
  const float* W4 = (const float*)d_in[17]; const float* as4 = (const float*)d_in[18];
  const float* ad4 = (const float*)d_in[19]; const float* b4 = (const float*)d_in[20];
  const float* p1 = (const float*)d_in[21]; const float* p2 = (const float*)d_in[22];

  const int E    = in_sizes[1] / 2;
  const int ETOT = E + NTN;

  // workspace carve-up
  char* ws = (char*)d_ws;
  size_t off = 0;
  auto carve = [&](size_t bytes) -> void* {
    void* pp = (void*)(ws + off);
    off += (bytes + 255) & ~(size_t)255;
    return pp;
  };
  float*    bufA = (float*)carve((size_t)NTN * 48 * 4);
  float*    bufB = (float*)carve((size_t)NTN * 48 * 4);
  float*    bufC = (float*)carve((size_t)NTN * 48 * 4);
  float*    ebuf = (float*)carve((size_t)ETOT * 3 * 4);
  float*    als  = (float*)carve((size_t)NTN * 3 * 4);
  float*    ald  = (float*)carve((size_t)NTN * 3 * 4);
  unsigned* mkey = (unsigned*)carve((size_t)NTN * 3 * 4);
  float*    den  = (float*)carve((size_t)NTN * 3 * 4);
  int*      nm1  = (int*)carve((size_t)NTN * 4);
  int*      nm2  = (int*)carve((size_t)NTN * 4);
  float*    mu   = (float*)carve(NFEAT * 4);
  float*    rstd = (float*)carve(NFEAT * 4);
  float*    W1f  = (float*)carve(NFEAT * 48 * 4);
  float*    brow = (float*)carve(64 * 4);
  float*    pn   = (float*)carve(64 * 4);

  union { float f; unsigned u; } cv; cv.f = NEGV;
  const unsigned negkey = (cv.u & 0x80000000u) ? ~cv.u : (cv.u | 0x80000000u);

  const int gemmBlocks = NTN / (16 * 8);          // 8 waves/block, 16 rows/wave
  const int eBlocks    = (ETOT + 255) / 256;

  // batchnorm folded into W1
  bn_stats_kernel<<<NFEAT, 256, 0, stream>>>(x, mu, rstd);
  fold_w1_kernel<<<1, 256, 0, stream>>>(W1, gamma, beta, mu, rstd, W1f, brow);

  // ---- layer 1: K=128 -> 48, H=3, C=16 ----
  wmma_gemm_kernel<128, 48><<<gemmBlocks, 256, 0, stream>>>(x, W1f, bufA, brow);
  attn_logits_kernel<<<(NTN * 3 + 255) / 256, 256, 0, stream>>>(bufA, as1, ad1, als, ald, 3, 16, NTN * 3);
  fill_u32_kernel<<<(NTN * 3 + 255) / 256, 256, 0, stream>>>(mkey, negkey, NTN * 3);
  fill_f32_kernel<<<(NTN * 3 + 255) / 256, 256, 0, stream>>>(den, 0.f, NTN * 3);
  init_bias_kernel<<<(NTN * 48 + 255) / 256, 256, 0, stream>>>(bufB, b1, 48, NTN * 48);
  edge_pass1_kernel<3><<<eBlocks, 256, 0, stream>>>(ei, E, ETOT, nullptr, als, ald, ebuf, mkey);
  edge_pass2_kernel<3><<<eBlocks, 256, 0, stream>>>(ei, E, ETOT, nullptr, ebuf, mkey, den);
  edge_pass3_kernel<3, 16><<<eBlocks, 256, 0, stream>>>(ei, E, ETOT, ebuf, den, bufA, bufB);

  // ---- layer 2: 48 -> 48, H=3, C=16 ----
  wmma_gemm_kernel<48, 48><<<gemmBlocks, 256, 0, stream>>>(bufB, W2, bufC, nullptr);
  attn_logits_kernel<<<(NTN * 3 + 255) / 256, 256, 0, stream>>>(bufC, as2, ad2, als, ald, 3, 16, NTN * 3);
  fill_u32_kernel<<<(NTN * 3 + 255) / 256, 256, 0, stream>>>(mkey, negkey, NTN * 3);
  fill_f32_kernel<<<(NTN * 3 + 255) / 256, 256, 0, stream>>>(den, 0.f, NTN * 3);
  init_bias_kernel<<<(NTN * 48 + 255) / 256, 256, 0, stream>>>(bufA, b2, 48, NTN * 48);
  edge_pass1_kernel<3><<<eBlocks, 256, 0, stream>>>(ei, E, ETOT, nullptr, als, ald, ebuf, mkey);
  edge_pass2_kernel<3><<<eBlocks, 256, 0, stream>>>(ei, E, ETOT, nullptr, ebuf, mkey, den);
  edge_pass3_kernel<3, 16><<<eBlocks, 256, 0, stream>>>(ei, E, ETOT, ebuf, den, bufC, bufA);

  // ---- pool 1: keep 1024/graph ----
  vec_norm_kernel<<<1, 64, 0, stream>>>(p1, 48, pn);
  topk_pool_kernel<48><<<BGR, 1024, 0, stream>>>(bufA, p1, pn, nullptr, nm1, bufB, NPERG / 2);

  // ---- layer 3: 48 -> 32, H=2, C=16 (edges masked by nm1) ----
  wmma_gemm_kernel<48, 32><<<gemmBlocks, 256, 0, stream>>>(bufB, W3, bufC, nullptr);
  attn_logits_kernel<<<(NTN * 2 + 255) / 256, 256, 0, stream>>>(bufC, as3, ad3, als, ald, 2, 16, NTN * 2);
  fill_u32_kernel<<<(NTN * 2 + 255) / 256, 256, 0, stream>>>(mkey, negkey, NTN * 2);
  fill_f32_kernel<<<(NTN * 2 + 255) / 256, 256, 0, stream>>>(den, 0.f, NTN * 2);
  init_bias_kernel<<<(NTN * 32 + 255) / 256, 256, 0, stream>>>(bufA, b3, 32, NTN * 32);
  edge_pass1_kernel<2><<<eBlocks, 256, 0, stream>>>(ei, E, ETOT, nm1, als, ald, ebuf, mkey);
  edge_pass2_kernel<2><<<eBlocks, 256, 0, stream>>>(ei, E, ETOT, nm1, ebuf, mkey, den);
  edge_pass3_kernel<2, 16><<<eBlocks, 256, 0, stream>>>(ei, E, ETOT, ebuf, den, bufC, bufA);

  // ---- layer 4: 32 -> 10, H=1, C=10 ----
  wmma_gemm_kernel<32, 10><<<gemmBlocks, 256, 0, stream>>>(bufA, W4, bufC, nullptr);
  attn_logits_kernel<<<(NTN + 255) / 256, 256, 0, stream>>>(bufC, as4, ad4, als, ald, 1, 10, NTN);
  fill_u32_kernel<<<(NTN + 255) / 256, 256, 0, stream>>>(mkey, negkey, NTN);
  fill_f32_kernel<<<(NTN + 255) / 256, 256, 0, stream>>>(den, 0.f, NTN);
  init_bias_kernel<<<(NTN * 10 + 255) / 256, 256, 0, stream>>>(bufB, b4, 10, NTN * 10);
  edge_pass1_kernel<1><<<eBlocks, 256, 0, stream>>>(ei, E, ETOT, nm1, als, ald, ebuf, mkey);
  edge_pass2_kernel<1><<<eBlocks, 256, 0, stream>>>(ei, E, ETOT, nm1, ebuf, mkey, den);
  edge_pass3_kernel<1, 10><<<eBlocks, 256, 0, stream>>>(ei, E, ETOT, ebuf, den, bufC, bufB);

  // ---- pool 2: keep 512/graph ----
  vec_norm_kernel<<<1, 64, 0, stream>>>(p2, 10, pn);
  topk_pool_kernel<10><<<BGR, 1024, 0, stream>>>(bufB, p2, pn, nm1, nm2, bufA, NPERG / 4);

  // ---- mean pool + log_softmax ----
  pooled_lsm_kernel<<<BGR, 256, 0, stream>>>(bufA, (float*)d_out);
}